// LMNNLoss_7146825581133
// MI455X (gfx1250) — compile-verified
//
#include <hip/hip_runtime.h>
#include <cstddef>
#include <cstdint>

typedef float v2f __attribute__((ext_vector_type(2)));
typedef float v8f __attribute__((ext_vector_type(8)));
typedef int   v4i __attribute__((ext_vector_type(4)));

#define MAX_N 4096

#if __has_builtin(__builtin_amdgcn_global_load_async_to_lds_b128) && \
    __has_builtin(__builtin_amdgcn_s_wait_asynccnt)
#define HAVE_ASYNC_LDS 1
typedef __attribute__((address_space(1))) v4i* gbl_v4i_p;   // global src
typedef __attribute__((address_space(3))) v4i* lds_v4i_p;   // LDS dst
__device__ __forceinline__ gbl_v4i_p to_gbl(const void* p) {
    return (gbl_v4i_p)(uintptr_t)p;
}
__device__ __forceinline__ lds_v4i_p to_lds(void* p) {
    // generic LDS pointers carry the LDS offset in the low 32 bits
    return (lds_v4i_p)(uint32_t)(uintptr_t)p;
}
#else
#define HAVE_ASYNC_LDS 0
#endif

// ---------------------------------------------------------------------------
// Kernel 1: row norms  x2[i] = sum_k X[i,k]^2
// ---------------------------------------------------------------------------
__global__ void row_norms_kernel(const float* __restrict__ X,
                                 float* __restrict__ x2,
                                 int n, int dfeat) {
    __shared__ float red[256];
    const int row = blockIdx.x;
    const float* xr = X + (size_t)row * dfeat;
    float s = 0.0f;
    for (int k = threadIdx.x; k < dfeat; k += blockDim.x) {
        float v = xr[k];
        s += v * v;
    }
    red[threadIdx.x] = s;
    __syncthreads();
    for (int st = 128; st > 0; st >>= 1) {
        if ((int)threadIdx.x < st) red[threadIdx.x] += red[threadIdx.x + st];
        __syncthreads();
    }
    if (threadIdx.x == 0) x2[row] = red[0];
}

// ---------------------------------------------------------------------------
// Kernel 2: D[i,j] = max(x2[i] + x2[j] - 2 * dot(X[i], X[j]), 0)
// Each wave computes a 32x32 macro-tile = 2x2 WMMA tiles with 4 independent
// accumulators (breaks the D->C serial chain, halves fragment loads:
// 4 b64 loads feed 4 v_wmma_f32_16x16x4_f32 per k-step).
// A fragment (16x4 fp32): lanes 0-15 hold row M=lane, K=k..k+1;
//                         lanes 16-31 hold row M=lane-16, K=k+2..k+3.
// B fragment (4x16) mirrors A (B = X[j-tile] rows over the same K slice).
// C/D (16x16 fp32): lane L, reg r -> M = r + 8*(L>>4), N = L&15.
// ---------------------------------------------------------------------------
__global__ void dist_gemm_kernel(const float* __restrict__ X,
                                 const float* __restrict__ x2,
                                 float* __restrict__ D,
                                 int n, int dfeat) {
    const int wave = threadIdx.x >> 5;
    const int lane = threadIdx.x & 31;
    const int macroDim = n >> 5;                       // 32-wide macro tiles
    const long totalMacro = (long)macroDim * macroDim;
    const long t = (long)blockIdx.x * (blockDim.x >> 5) + wave;
    if (t >= totalMacro) return;   // whole wave exits: EXEC stays all-1s

    const int ti = (int)(t / macroDim);
    const int tj = (int)(t % macroDim);
    const int i0 = ti << 5;
    const int j0 = tj << 5;

    const int r  = lane & 15;   // row within 16-tile for this lane's fragment
    const int kp = lane >> 4;   // K-pair select (0 -> k..k+1, 1 -> k+2..k+3)

    const float* __restrict__ a0p = X + (size_t)(i0 + r) * dfeat + 2 * kp;
    const float* __restrict__ a1p = a0p + (size_t)16 * dfeat;
    const float* __restrict__ b0p = X + (size_t)(j0 + r) * dfeat + 2 * kp;
    const float* __restrict__ b1p = b0p + (size_t)16 * dfeat;

    v8f acc00 = {0.f,0.f,0.f,0.f,0.f,0.f,0.f,0.f};
    v8f acc01 = {0.f,0.f,0.f,0.f,0.f,0.f,0.f,0.f};
    v8f acc10 = {0.f,0.f,0.f,0.f,0.f,0.f,0.f,0.f};
    v8f acc11 = {0.f,0.f,0.f,0.f,0.f,0.f,0.f,0.f};

#if __has_builtin(__builtin_amdgcn_wmma_f32_16x16x4_f32)
#pragma unroll 4
    for (int k = 0; k < dfeat; k += 4) {
        v2f a0 = *(const v2f*)(a0p + k);
        v2f a1 = *(const v2f*)(a1p + k);
        v2f b0 = *(const v2f*)(b0p + k);
        v2f b1 = *(const v2f*)(b1p + k);
        acc00 = __builtin_amdgcn_wmma_f32_16x16x4_f32(false, a0, false, b0,
                                                      (short)0, acc00, false, false);
        acc01 = __builtin_amdgcn_wmma_f32_16x16x4_f32(false, a0, false, b1,
                                                      (short)0, acc01, false, false);
        acc10 = __builtin_amdgcn_wmma_f32_16x16x4_f32(false, a1, false, b0,
                                                      (short)0, acc10, false, false);
        acc11 = __builtin_amdgcn_wmma_f32_16x16x4_f32(false, a1, false, b1,
                                                      (short)0, acc11, false, false);
    }
#else
    // Scalar fallback (should not be needed on gfx1250)
    for (int rr = 0; rr < 8; ++rr) {
        const int mi = rr + 8 * kp;
        float d00 = 0.f, d01 = 0.f, d10 = 0.f, d11 = 0.f;
        for (int k = 0; k < dfeat; ++k) {
            const float xi0 = X[(size_t)(i0 + mi) * dfeat + k];
            const float xi1 = X[(size_t)(i0 + 16 + mi) * dfeat + k];
            const float xj0 = X[(size_t)(j0 + r) * dfeat + k];
            const float xj1 = X[(size_t)(j0 + 16 + r) * dfeat + k];
            d00 += xi0 * xj0; d01 += xi0 * xj1;
            d10 += xi1 * xj0; d11 += xi1 * xj1;
        }
        acc00[rr] = d00; acc01[rr] = d01; acc10[rr] = d10; acc11[rr] = d11;
    }
#endif

    // Epilogue: dot products -> squared distances for the 4 sub-tiles
    const int gj0 = j0 + r;
    const int gj1 = j0 + 16 + r;
    const float xj0 = x2[gj0];
    const float xj1 = x2[gj1];
#pragma unroll
    for (int rr = 0; rr < 8; ++rr) {
        const int gi0 = i0 + rr + 8 * kp;
        const int gi1 = gi0 + 16;
        const float xi0 = x2[gi0];
        const float xi1 = x2[gi1];
        D[(size_t)gi0 * n + gj0] = fmaxf(xi0 + xj0 - 2.0f * acc00[rr], 0.0f);
        D[(size_t)gi0 * n + gj1] = fmaxf(xi0 + xj1 - 2.0f * acc01[rr], 0.0f);
        D[(size_t)gi1 * n + gj0] = fmaxf(xi1 + xj0 - 2.0f * acc10[rr], 0.0f);
        D[(size_t)gi1 * n + gj1] = fmaxf(xi1 + xj1 - 2.0f * acc11[rr], 0.0f);
    }
}

// ---------------------------------------------------------------------------
// Top-3 helper
// ---------------------------------------------------------------------------
__device__ __forceinline__ void top3_insert(float v, int idx,
                                            float bv[3], int bi[3]) {
    if (v < bv[2]) {
        if (v < bv[1]) {
            bv[2] = bv[1]; bi[2] = bi[1];
            if (v < bv[0]) {
                bv[1] = bv[0]; bi[1] = bi[0];
                bv[0] = v; bi[0] = idx;
            } else {
                bv[1] = v; bi[1] = idx;
            }
        } else {
            bv[2] = v; bi[2] = idx;
        }
    }
}

// ---------------------------------------------------------------------------
// Kernel 3: per-row same-class 3-NN (excluding self).
// ---------------------------------------------------------------------------
#define TK_T 128
__global__ void topk_kernel(const float* __restrict__ D,
                            const int* __restrict__ labels,
                            float* __restrict__ radii,
                            float* __restrict__ pullArr,
                            int* __restrict__ tn,
                            int n) {
    __shared__ float sv[TK_T * 3];
    __shared__ int   si[TK_T * 3];
    const int i   = blockIdx.x;
    const int tid = threadIdx.x;
    const int li  = labels[i];

    const float INF = __builtin_inff();
    float bv[3] = {INF, INF, INF};
    int   bi[3] = {-1, -1, -1};

    const float* __restrict__ drow = D + (size_t)i * n;
    for (int j = tid; j < n; j += TK_T) {
        if (j == i || labels[j] != li) continue;
        top3_insert(drow[j], j, bv, bi);
    }
    sv[tid * 3 + 0] = bv[0]; si[tid * 3 + 0] = bi[0];
    sv[tid * 3 + 1] = bv[1]; si[tid * 3 + 1] = bi[1];
    sv[tid * 3 + 2] = bv[2]; si[tid * 3 + 2] = bi[2];
    __syncthreads();

    for (int s = TK_T / 2; s > 0; s >>= 1) {
        if (tid < s) {
            const int o = (tid + s) * 3;
#pragma unroll
            for (int t = 0; t < 3; ++t) top3_insert(sv[o + t], si[o + t], bv, bi);
            sv[tid * 3 + 0] = bv[0]; si[tid * 3 + 0] = bi[0];
            sv[tid * 3 + 1] = bv[1]; si[tid * 3 + 1] = bi[1];
            sv[tid * 3 + 2] = bv[2]; si[tid * 3 + 2] = bi[2];
        }
        __syncthreads();
    }

    if (tid == 0) {
        radii[i]   = 1.0f + bv[2];
        pullArr[i] = bv[0] + bv[1] + bv[2];
        tn[i * 3 + 0] = bi[0];
        tn[i * 3 + 1] = bi[1];
        tn[i * 3 + 2] = bi[2];
    }
}

// ---------------------------------------------------------------------------
// Kernel 4: bitonic sort radii (ascending) + suffix sums.
// ---------------------------------------------------------------------------
__global__ void sort_scan_kernel(const float* __restrict__ radii,
                                 float* __restrict__ rs,
                                 float* __restrict__ ssum,
                                 int n) {
    __shared__ float s[MAX_N];
    __shared__ float tmp[MAX_N];
    const int tid = threadIdx.x;
    const int NT  = blockDim.x;
    const float INF = __builtin_inff();

    for (int i = tid; i < MAX_N; i += NT) s[i] = (i < n) ? radii[i] : INF;
    __syncthreads();

    for (int k = 2; k <= MAX_N; k <<= 1) {
        for (int j = k >> 1; j > 0; j >>= 1) {
            for (int i = tid; i < MAX_N; i += NT) {
                const int ixj = i ^ j;
                if (ixj > i) {
                    const bool up = ((i & k) == 0);
                    const float a = s[i];
                    const float b = s[ixj];
                    if (up ? (a > b) : (a < b)) { s[i] = b; s[ixj] = a; }
                }
            }
            __syncthreads();
        }
    }

    for (int i = tid; i < n; i += NT) rs[i] = s[i];
    __syncthreads();

    for (int i = tid; i < MAX_N; i += NT) if (i >= n) s[i] = 0.0f;
    __syncthreads();

    float* in  = s;
    float* out = tmp;
    for (int off = 1; off < MAX_N; off <<= 1) {
        for (int i = tid; i < MAX_N; i += NT)
            out[i] = in[i] + ((i >= off) ? in[i - off] : 0.0f);
        __syncthreads();
        float* t2 = in; in = out; out = t2;
    }
    const float total = in[n - 1];
    for (int i = tid; i < n; i += NT)
        ssum[i] = total - ((i > 0) ? in[i - 1] : 0.0f);
    if (tid == 0) ssum[n] = 0.0f;
}

// ---------------------------------------------------------------------------
// Kernel 5: push term. One block (256 threads) per row `a`.
//   contribution(d) = ssum[pos] - d * (n - pos),  pos = upper_bound(rs, d)
// rs/ssum staged into LDS via GLOBAL_LOAD_ASYNC_TO_LDS_B128 (ASYNCcnt path).
// ---------------------------------------------------------------------------
__global__ void push_kernel(const float* __restrict__ D,
                            const float* __restrict__ rs,
                            const float* __restrict__ ssum,
                            const int* __restrict__ tn,
                            float* __restrict__ partials,
                            int n) {
    __shared__ float lrs[MAX_N];
    __shared__ float lss[MAX_N + 4];
    __shared__ float red[256];
    const int a   = blockIdx.x;
    const int tid = threadIdx.x;

#if HAVE_ASYNC_LDS
    {
        const int c_rs = n >> 2;              // 16-byte chunks of rs
        const int c_ss = (n + 4) >> 2;        // covers n+1 entries
        for (int c = tid; c < c_rs; c += 256)
            __builtin_amdgcn_global_load_async_to_lds_b128(
                to_gbl(rs + 4 * c), to_lds(lrs + 4 * c), 0, 0);
        for (int c = tid; c < c_ss; c += 256)
            __builtin_amdgcn_global_load_async_to_lds_b128(
                to_gbl(ssum + 4 * c), to_lds(lss + 4 * c), 0, 0);
        __builtin_amdgcn_s_wait_asynccnt(0);
    }
#else
    for (int i = tid; i < n; i += 256) lrs[i] = rs[i];
    for (int i = tid; i <= n; i += 256) lss[i] = ssum[i];
#endif
    __syncthreads();

    const int t0 = tn[a * 3 + 0];
    const int t1 = tn[a * 3 + 1];
    const int t2 = tn[a * 3 + 2];

    const float* __restrict__ drow = D + (size_t)a * n;
    float local = 0.0f;
    for (int j = tid; j < n; j += 256) {
        if (j == a || j == t0 || j == t1 || j == t2) continue;
        const float d = drow[j];
        int lo = 0, hi = n;
        while (lo < hi) {                 // first index with rs[idx] > d
            const int m = (lo + hi) >> 1;
            if (lrs[m] <= d) lo = m + 1; else hi = m;
        }
        local += lss[lo] - d * (float)(n - lo);
    }
    red[tid] = local;
    __syncthreads();
    for (int st = 128; st > 0; st >>= 1) {
        if (tid < st) red[tid] += red[tid + st];
        __syncthreads();
    }
    if (tid == 0) partials[a] = red[0];
}

// ---------------------------------------------------------------------------
// Kernel 6: final deterministic reduction -> out[0] = (pull + push) / n
// ---------------------------------------------------------------------------
__global__ void final_kernel(const float* __restrict__ pullArr,
                             const float* __restrict__ partials,
                             float* __restrict__ out,
                             int n) {
    __shared__ float red[256];
    const int tid = threadIdx.x;
    float s = 0.0f;
    for (int i = tid; i < n; i += 256) s += pullArr[i] + partials[i];
    red[tid] = s;
    __syncthreads();
    for (int st = 128; st > 0; st >>= 1) {
        if (tid < st) red[tid] += red[tid + st];
        __syncthreads();
    }
    if (tid == 0) out[0] = red[0] / (float)n;
}

// ---------------------------------------------------------------------------
// Launch
// ---------------------------------------------------------------------------
extern "C" void kernel_launch(void* const* d_in, const int* in_sizes, int n_in,
                              void* d_out, int out_size, void* d_ws, size_t ws_size,
                              hipStream_t stream) {
    const float* X      = (const float*)d_in[0];
    const int*   labels = (const int*)d_in[1];
    const int n     = in_sizes[1];            // 4096
    const int dfeat = in_sizes[0] / n;        // 256

    // workspace layout (floats)
    float* f        = (float*)d_ws;
    const size_t nD = (size_t)n * n;
    float* Dm       = f;                      // n*n   distance matrix (64 MB)
    float* x2       = Dm + nD;                // n     row norms
    float* radii    = x2 + n;                 // n
    float* rs       = radii + n;              // n     sorted radii
    float* ssum     = rs + n;                 // n+4   suffix sums (padded)
    float* pullArr  = ssum + n + 4;           // n
    float* partials = pullArr + n;            // n
    int*   tn       = (int*)(partials + n);   // n*3   target-neighbor indices

    float* out = (float*)d_out;

    // 1) row norms
    row_norms_kernel<<<n, 256, 0, stream>>>(X, x2, n, dfeat);

    // 2) WMMA distance GEMM: one 32x32 macro-tile per wave, 8 waves/block
    {
        const int macroDim = n / 32;
        const long tiles   = (long)macroDim * macroDim;
        const int wavesPerBlock = 8;
        const int blocks = (int)((tiles + wavesPerBlock - 1) / wavesPerBlock);
        dist_gemm_kernel<<<blocks, wavesPerBlock * 32, 0, stream>>>(X, x2, Dm, n, dfeat);
    }

    // 3) per-row same-class top-3
    topk_kernel<<<n, TK_T, 0, stream>>>(Dm, labels, radii, pullArr, tn, n);

    // 4) sort radii + suffix sums (single block)
    sort_scan_kernel<<<1, 1024, 0, stream>>>(radii, rs, ssum, n);

    // 5) push term, one block per row (async-LDS staged binary search tables)
    push_kernel<<<n, 256, 0, stream>>>(Dm, rs, ssum, tn, partials, n);

    // 6) final scalar
    final_kernel<<<1, 256, 0, stream>>>(pullArr, partials, out, n);
}